// LSM_66838281061160
// MI455X (gfx1250) — compile-verified
//
#include <hip/hip_runtime.h>
#include <hip/hip_bf16.h>

typedef __attribute__((ext_vector_type(2))) float v2f;
typedef __attribute__((ext_vector_type(8))) float v8f;

#define LSM_N 8192
#define LSM_D 64
#define TJ 4                      // j-tiles per wave
#define LSM_EPS 1e-6f

__device__ __forceinline__ float wave_sum(float v) {
#pragma unroll
  for (int off = 16; off > 0; off >>= 1) v += __shfl_xor(v, off, 32);
  return v;
}

// ---------------- Kernel 1: per-row sum and sum-of-squares of Z ------------
__global__ void __launch_bounds__(256)
lsm_rowstats(const float* __restrict__ Z, float* __restrict__ sq,
             float* __restrict__ s) {
  const int row  = blockIdx.x * 8 + (threadIdx.x >> 5);   // one wave per row
  const int lane = threadIdx.x & 31;
  const float z0 = Z[row * LSM_D + lane];
  const float z1 = Z[row * LSM_D + lane + 32];
  float q = z0 * z0 + z1 * z1;
  float t = z0 + z1;
  q = wave_sum(q);
  t = wave_sum(t);
  if (lane == 0) { sq[row] = q; s[row] = t; }
}

// ---------------- Kernel 2: fused WMMA gram + elementwise LL ---------------
__global__ void __launch_bounds__(256)
lsm_main(const int* __restrict__ A, const float* __restrict__ alpha,
         const float* __restrict__ Z, const float* __restrict__ sq,
         const float* __restrict__ s, float* __restrict__ partials) {
  const int wave = threadIdx.x >> 5;
  const int lane = threadIdx.x & 31;
  const int half = lane >> 4;   // selects K-pair (A/B frag) and M-half (C/D)
  const int lm   = lane & 15;   // M for A-frag, N for B-frag / C rows

  const int i0     = blockIdx.x * 16;
  const int j_base = blockIdx.y * (8 * TJ * 16) + wave * (TJ * 16);

  // Preload the 16x64 A-operand (rows i0..i0+15 of Z), ISA 16x4 f32 layout:
  // lane = {half,lm}: M = lm, K = kk*4 + half*2 + {0,1}. Reused for all TJ tiles.
  v2f afrag[16];
  {
    const int ia = i0 + lm;
#pragma unroll
    for (int kk = 0; kk < 16; ++kk) {
      const int k = kk * 4 + half * 2;
      afrag[kk].x = Z[ia * LSM_D + k];
      afrag[kk].y = Z[ia * LSM_D + k + 1];
    }
  }

  // Row-side scalars for the 8 accumulator rows this lane owns.
  float sqi[8], si[8];
#pragma unroll
  for (int r = 0; r < 8; ++r) {
    const int i = i0 + r + half * 8;   // C/D layout: VGPR r -> M = r + half*8
    sqi[r] = sq[i];
    si[r]  = s[i];
  }

  float accA = 0.f, accS = 0.f, accDiag = 0.f;

  for (int jt = 0; jt < TJ; ++jt) {
    const int j0 = j_base + jt * 16;
    const int j  = j0 + lm;            // this lane's column index

    // warm the A strip for this tile
    __builtin_prefetch(&A[(size_t)(i0 + lm) * LSM_N + j0], 0, 0);

    // gram tile: D = A x B (+C), K=64 in 16 steps of 4
    v8f c = {};
#pragma unroll
    for (int kk = 0; kk < 16; ++kk) {
      v2f b;
      const int k = kk * 4 + half * 2;
      b.x = Z[j * LSM_D + k];
      b.y = Z[j * LSM_D + k + 1];
      c = __builtin_amdgcn_wmma_f32_16x16x4_f32(false, afrag[kk], false, b,
                                                (short)0, c, false, false);
    }

    const float sqj = sq[j];
    const float sj  = s[j];
    const float aj  = alpha[j];

#pragma unroll
    for (int r = 0; r < 8; ++r) {
      const int i   = i0 + r + half * 8;
      const float g = c[r];
      float d2 = sqi[r] + sqj - 2.f * g
               + (2.f * LSM_EPS) * (si[r] - sj)
               + (float)LSM_D * LSM_EPS * LSM_EPS;
      const float z  = sqrtf(fmaxf(d2, 0.f));
      const float th = aj - z;
      // numerically stable softplus
      const float sp = fmaxf(th, 0.f) + __logf(1.f + __expf(-fabsf(th)));
      const int aij  = A[(size_t)i * LSM_N + j];
      accA += th * (float)aij;
      accS += sp;
      if (i == j) accDiag += sp;
    }
  }

  accA    = wave_sum(accA);
  accS    = wave_sum(accS);
  accDiag = wave_sum(accDiag);

  __shared__ float redA[8], redS[8], redD[8];
  if (lane == 0) { redA[wave] = accA; redS[wave] = accS; redD[wave] = accDiag; }
  __syncthreads();
  if (threadIdx.x == 0) {
    float tA = 0.f, tS = 0.f, tD = 0.f;
#pragma unroll
    for (int w = 0; w < 8; ++w) { tA += redA[w]; tS += redS[w]; tD += redD[w]; }
    const int bid = blockIdx.y * gridDim.x + blockIdx.x;
    partials[bid * 3 + 0] = tA;
    partials[bid * 3 + 1] = tS;
    partials[bid * 3 + 2] = tD;
  }
}

// ---------------- Kernel 3: deterministic final reduction ------------------
__global__ void __launch_bounds__(256)
lsm_reduce(const float* __restrict__ partials, int nPart,
           float* __restrict__ out) {
  __shared__ float sA[256], sS[256], sD[256];
  float tA = 0.f, tS = 0.f, tD = 0.f;
  for (int p = threadIdx.x; p < nPart; p += 256) {
    tA += partials[p * 3 + 0];
    tS += partials[p * 3 + 1];
    tD += partials[p * 3 + 2];
  }
  sA[threadIdx.x] = tA; sS[threadIdx.x] = tS; sD[threadIdx.x] = tD;
  __syncthreads();
  for (int stride = 128; stride > 0; stride >>= 1) {
    if (threadIdx.x < stride) {
      sA[threadIdx.x] += sA[threadIdx.x + stride];
      sS[threadIdx.x] += sS[threadIdx.x + stride];
      sD[threadIdx.x] += sD[threadIdx.x + stride];
    }
    __syncthreads();
  }
  if (threadIdx.x == 0)
    out[0] = 0.5f * sA[0] - 0.5f * (sS[0] - sD[0]);
}

extern "C" void kernel_launch(void* const* d_in, const int* in_sizes, int n_in,
                              void* d_out, int out_size, void* d_ws, size_t ws_size,
                              hipStream_t stream) {
  const int*   A     = (const int*)d_in[0];     // [N*N] int32
  const float* alpha = (const float*)d_in[1];   // [N]
  const float* Z     = (const float*)d_in[2];   // [N*D]
  float* ws       = (float*)d_ws;
  float* sq       = ws;                 // [N]
  float* s        = ws + LSM_N;         // [N]
  float* partials = ws + 2 * LSM_N;     // [nBlocks*3]
  float* out      = (float*)d_out;

  lsm_rowstats<<<LSM_N / 8, 256, 0, stream>>>(Z, sq, s);

  dim3 grid(LSM_N / 16, LSM_N / (8 * TJ * 16));   // 512 x 16
  lsm_main<<<grid, 256, 0, stream>>>(A, alpha, Z, sq, s, partials);

  const int nPart = (LSM_N / 16) * (LSM_N / (8 * TJ * 16));
  lsm_reduce<<<1, 256, 0, stream>>>(partials, nPart, out);
}